// RegionalConv1Layer_34651796144297
// MI455X (gfx1250) — compile-verified
//
#include <hip/hip_runtime.h>
#include <hip/hip_bf16.h>
#include <math.h>

typedef __attribute__((ext_vector_type(2))) float v2f;
typedef __attribute__((ext_vector_type(8))) float v8f;

#define B_SZ   256
#define T_SZ   8192
#define CIN    6
#define COUT   8
#define KS     10
#define KTOT   60            // KS*CIN
#define TILE_T 16
#define WAVES  8
#define TPW    4             // tiles per wave
#define WG_T   (TILE_T * WAVES * TPW)   // 512 timesteps per workgroup
#define NCHUNK 15            // K chunks of 4 covering K=0..59

// workspace layout (floats)
#define WS_STATS 0                        // 256 * 12   (mean0..5, std0..5)
#define WS_BPK   4096                     // 256 * 960  (15 chunks x 32 lanes x v2f)
#define WS_BEFF  (4096 + 245760)          // 256 * 8
#define WS_VPAD  (4096 + 245760 + 2048)   // 256 * 6

// ---------------- Kernel 1: per-batch channel statistics -------------------
__global__ __launch_bounds__(256) void stats_kernel(const float* __restrict__ x,
                                                    float* __restrict__ stats) {
  const int b = blockIdx.x, tid = threadIdx.x;
  const float* xb = x + (size_t)b * T_SZ * CIN;
  float s[CIN] = {}, ss[CIN] = {};
  for (int t = tid; t < T_SZ; t += 256) {
    const v2f* p = (const v2f*)(xb + t * CIN);   // row stride 24B -> 8B aligned
    v2f d0 = p[0], d1 = p[1], d2 = p[2];
    s[0] += d0.x; ss[0] += d0.x * d0.x;
    s[1] += d0.y; ss[1] += d0.y * d0.y;
    s[2] += d1.x; ss[2] += d1.x * d1.x;
    s[3] += d1.y; ss[3] += d1.y * d1.y;
    s[4] += d2.x; ss[4] += d2.x * d2.x;
    s[5] += d2.y; ss[5] += d2.y * d2.y;
  }
  __shared__ float red[256][12];
#pragma unroll
  for (int i = 0; i < CIN; ++i) { red[tid][i] = s[i]; red[tid][6 + i] = ss[i]; }
  __syncthreads();
  for (int off = 128; off > 0; off >>= 1) {
    if (tid < off)
#pragma unroll
      for (int j = 0; j < 12; ++j) red[tid][j] += red[tid + off][j];
    __syncthreads();
  }
  if (tid == 0) {
#pragma unroll
    for (int i = 0; i < CIN; ++i) {
      float mean = red[0][i] * (1.0f / T_SZ);
      float var  = red[0][6 + i] * (1.0f / T_SZ) - mean * mean;
      stats[b * 12 + i]     = mean;
      stats[b * 12 + 6 + i] = sqrtf(fmaxf(var, 0.0f));
    }
  }
}

// ------- Kernel 2: hypernetwork + fold normalization into conv weights -----
__global__ __launch_bounds__(256) void hyper_kernel(
    const float* __restrict__ attrs,
    const float* __restrict__ bw1, const float* __restrict__ bb1,
    const float* __restrict__ bw2, const float* __restrict__ bb2,
    const float* __restrict__ ww1, const float* __restrict__ wb1,
    const float* __restrict__ ww2, const float* __restrict__ wb2,
    const float* __restrict__ iw1, const float* __restrict__ ib1,
    const float* __restrict__ iw2, const float* __restrict__ ib2,
    const float* __restrict__ stats,
    float* __restrict__ Bpk, float* __restrict__ beff, float* __restrict__ vpad) {
  const int b = blockIdx.x, tid = threadIdx.x;
  __shared__ float h[3][8];
  __shared__ float kb[400], kw[80], kbias[8];
  const float* a = attrs + b * 27;

  if (tid < 24) {  // 3 nets x 8 hidden units
    const int net = tid >> 3, j = tid & 7;
    const float* w1 = (net == 0) ? bw1 : (net == 1) ? ww1 : iw1;
    const float* b1 = (net == 0) ? bb1 : (net == 1) ? wb1 : ib1;
    float acc = b1[j];
#pragma unroll
    for (int i = 0; i < 27; ++i) acc += a[i] * w1[i * 8 + j];
    h[net][j] = fmaxf(acc, 0.0f);
  }
  __syncthreads();

  for (int idx = tid; idx < 400; idx += 256) {
    float acc = bb2[idx];
#pragma unroll
    for (int r = 0; r < 8; ++r) acc += h[0][r] * bw2[r * 400 + idx];
    kb[idx] = tanhf(acc);
  }
  if (tid < 80) {
    float acc = wb2[tid];
#pragma unroll
    for (int r = 0; r < 8; ++r) acc += h[1][r] * ww2[r * 80 + tid];
    kw[tid] = tanhf(acc);
  }
  if (tid < 8) {
    float acc = ib2[tid];
#pragma unroll
    for (int r = 0; r < 8; ++r) acc += h[2][r] * iw2[r * 8 + tid];
    kbias[tid] = tanhf(acc);
  }
  __syncthreads();

  const float* st = stats + b * 12;   // [mean0..5, std0..5]; idx 5 = q channel
  const float q_mean = st[5], q_std = st[11];

  // Pre-pack the per-lane WMMA B-operand register image:
  //   Bpk[b][c][lane] = { keff[4c+2*(lane>>4)][lane&15], keff[4c+2*(lane>>4)+1][lane&15] }
  // where keff[k][o] = a_i * kern[w,i,o] (k = w*6+i), zero-padded for k>=60 or o>=8.
  for (int idx = tid; idx < NCHUNK * 32; idx += 256) {
    const int c = idx >> 5, lane = idx & 31;
    const int m = lane & 15, khf = lane >> 4;
    const int k0 = 4 * c + 2 * khf;
    float v[2];
#pragma unroll
    for (int u = 0; u < 2; ++u) {
      const int k = k0 + u;
      float val = 0.0f;
      if (k < KTOT && m < COUT) {
        const int w = k / CIN, i = k - w * CIN;
        const float kern = (i < 5) ? kb[w * 40 + i * 8 + m] : kw[w * 8 + m];
        const float ai = (i < 5) ? (q_std / st[6 + i]) : 1.0f;
        val = ai * kern;
      }
      v[u] = val;
    }
    Bpk[(size_t)b * (NCHUNK * 64) + idx * 2]     = v[0];
    Bpk[(size_t)b * (NCHUNK * 64) + idx * 2 + 1] = v[1];
  }
  // Folded bias: beff[o] = bias[o] + sum_{w,i<5} c_i * k[w,i,o]
  if (tid < 8) {
    float acc = kbias[tid];
    for (int w = 0; w < KS; ++w)
#pragma unroll
      for (int i = 0; i < 5; ++i) {
        const float ai = q_std / st[6 + i];
        const float ci = q_mean - st[i] * ai;
        acc += ci * kb[w * 40 + i * 8 + tid];
      }
    beff[b * 8 + tid] = acc;
  }
  // Virtual left-pad input value so folded conv contributes exactly 0: v_i = -c_i/a_i
  if (tid < 6) {
    float v = 0.0f;
    if (tid < 5) v = st[tid] - q_mean * st[6 + tid] / q_std;
    vpad[b * 6 + tid] = v;
  }
}

// --------------- Kernel 3: WMMA causal conv + bias + ELU -------------------
__global__ __launch_bounds__(256) void conv_wmma_kernel(
    const float* __restrict__ x, const float* __restrict__ Bpk,
    const float* __restrict__ beff, const float* __restrict__ vpad,
    float* __restrict__ out) {
  const int b   = blockIdx.y;
  const int t0  = blockIdx.x * WG_T;
  const int tid = threadIdx.x;

  // Stage input window [t0-9, t0+WG_T) x 6ch into LDS, row-wise (3x b64 per row).
  __shared__ float sIn[(WG_T + KS - 1) * CIN];   // 521*6 = 3126 floats (12.5 KB)
  {
    const float* xb = x + (size_t)b * T_SZ * CIN;
    const float* vp = vpad + b * CIN;
    for (int r = tid; r < WG_T + KS - 1; r += 256) {
      const int t = t0 - (KS - 1) + r;
      v2f d0, d1, d2;
      if (t >= 0) {
        const v2f* src = (const v2f*)(xb + t * CIN);
        d0 = src[0]; d1 = src[1]; d2 = src[2];
      } else {
        d0.x = vp[0]; d0.y = vp[1];
        d1.x = vp[2]; d1.y = vp[3];
        d2.x = vp[4]; d2.y = vp[5];
      }
      v2f* dst = (v2f*)(sIn + r * CIN);
      dst[0] = d0; dst[1] = d1; dst[2] = d2;
    }
  }
  __syncthreads();

  const int lane = tid & 31, wv = tid >> 5;
  const int m  = lane & 15;       // M row (time) for A; N col for B/C
  const int kh = lane >> 4;       // which K-half of each chunk this lane holds

  // B operands: one coalesced b64 load per chunk (pre-packed per-lane image).
  const v2f* Bt = (const v2f*)(Bpk + (size_t)b * (NCHUNK * 64));
  v2f Bop[NCHUNK];
#pragma unroll
  for (int c = 0; c < NCHUNK; ++c) Bop[c] = Bt[c * 32 + lane];

  const float bo = beff[b * 8 + (m & 7)];   // valid for lanes with m<8

#pragma unroll
  for (int it = 0; it < TPW; ++it) {
    const int tile = it * WAVES + wv;                 // 0..31
    // A[mrow][k] = sIn[(tile*16 + mrow)*6 + k] : linear in k -> contiguous b64
    const float* abase = sIn + (tile * TILE_T + m) * CIN;

    v8f acc = {};
#pragma unroll
    for (int c = 0; c < NCHUNK; ++c) {
      const int k0 = 4 * c + 2 * kh;
      v2f A;
      A.x = abase[k0];
      A.y = abase[k0 + 1];
      acc = __builtin_amdgcn_wmma_f32_16x16x4_f32(
          /*neg_a=*/false, A, /*neg_b=*/false, Bop[c],
          /*c_mod=*/(short)0, acc, /*reuse_a=*/false, /*reuse_b=*/false);
    }

    // C VGPR j -> M = j + 8*kh, N = m. Valid outputs: N < 8.
    if (m < COUT) {
      float* ob = out + ((size_t)b * T_SZ + (t0 + tile * TILE_T + kh * 8)) * COUT + m;
#pragma unroll
      for (int j = 0; j < 8; ++j) {
        float v = acc[j] + bo;
        v = (v > 0.0f) ? v : expm1f(v);     // ELU(alpha=1)
        ob[(size_t)j * COUT] = v;
      }
    }
  }
}

extern "C" void kernel_launch(void* const* d_in, const int* in_sizes, int n_in,
                              void* d_out, int out_size, void* d_ws, size_t ws_size,
                              hipStream_t stream) {
  const float* x    = (const float*)d_in[0];
  const float* attrs= (const float*)d_in[1];
  const float* bw1  = (const float*)d_in[2];
  const float* bb1  = (const float*)d_in[3];
  const float* bw2  = (const float*)d_in[4];
  const float* bb2  = (const float*)d_in[5];
  const float* ww1  = (const float*)d_in[6];
  const float* wb1  = (const float*)d_in[7];
  const float* ww2  = (const float*)d_in[8];
  const float* wb2  = (const float*)d_in[9];
  const float* iw1  = (const float*)d_in[10];
  const float* ib1  = (const float*)d_in[11];
  const float* iw2  = (const float*)d_in[12];
  const float* ib2  = (const float*)d_in[13];
  float* out = (float*)d_out;

  float* ws    = (float*)d_ws;
  float* stats = ws + WS_STATS;
  float* bpk   = ws + WS_BPK;
  float* beff  = ws + WS_BEFF;
  float* vpad  = ws + WS_VPAD;

  stats_kernel<<<B_SZ, 256, 0, stream>>>(x, stats);
  hyper_kernel<<<B_SZ, 256, 0, stream>>>(attrs, bw1, bb1, bw2, bb2,
                                         ww1, wb1, ww2, wb2,
                                         iw1, ib1, iw2, ib2,
                                         stats, bpk, beff, vpad);
  dim3 grid(T_SZ / WG_T, B_SZ);
  conv_wmma_kernel<<<grid, 256, 0, stream>>>(x, bpk, beff, vpad, out);
}